// ETF_58806692217186
// MI455X (gfx1250) — compile-verified
//
#include <hip/hip_runtime.h>
#include <math.h>

#define H 512
#define W 512
#define HW (H * W)
#define PW 516                 // padded width/height (2-wide reflect halo)
#define PHW (PW * PW)
#define TD 36                  // ETF halo tile: 32 + 2*2

typedef unsigned u32x4 __attribute__((ext_vector_type(4)));
typedef int      i32x8 __attribute__((ext_vector_type(8)));
typedef int      i32x4 __attribute__((ext_vector_type(4)));

// ---------------------------------------------------------------------------
// CDNA5 async global->LDS copy (per-lane, ASYNCcnt)
// ---------------------------------------------------------------------------
__device__ __forceinline__ void async_g2l_b32(void* lds_dst, const float* gsrc) {
    asm volatile("global_load_async_to_lds_b32 %0, %1, off"
                 :
                 : "v"((unsigned)(size_t)lds_dst),
                   "v"((unsigned long long)(size_t)gsrc)
                 : "memory");
}
__device__ __forceinline__ void wait_async() {
    asm volatile("s_wait_asynccnt 0x0" ::: "memory");
}
__device__ __forceinline__ void wait_tensor() {
#if __has_builtin(__builtin_amdgcn_s_wait_tensorcnt)
    __builtin_amdgcn_s_wait_tensorcnt(0);
#else
    asm volatile("s_wait_tensorcnt 0x0" ::: "memory");
#endif
}

// ---------------------------------------------------------------------------
// CDNA5 Tensor Data Mover: 2D tile (tile_w x tile_h elems, fp32) from a
// row-major global array (row stride = stride elems) into contiguous LDS.
// D# packing per CDNA5 ISA ch.8 (groups 0/1; groups 2/3 zero for 2D).
// ---------------------------------------------------------------------------
#if __has_builtin(__builtin_amdgcn_tensor_load_to_lds)
#define HAVE_TDM 1
__device__ __forceinline__ void tdm_load_2d(void* lds_dst, const float* gsrc,
                                            unsigned tile_w, unsigned tile_h,
                                            unsigned stride) {
    const unsigned long long ga = (unsigned long long)(size_t)gsrc;
    u32x4 g0;
    g0.x = 1u;                                   // count=1, user mode, no gather
    g0.y = (unsigned)(size_t)lds_dst;            // lds_addr (byte offset)
    g0.z = (unsigned)ga;                         // global_addr[31:0]
    g0.w = (unsigned)(ga >> 32) | (2u << 30);    // global_addr[56:32] | type=2
    i32x8 g1;
    g1[0] = (int)(2u << 16);                     // wg_mask=0, data_size=4B, no flags
    g1[1] = (int)((unsigned)PW << 16);           // tensor_dim0[15:0] << 16
    g1[2] = (int)((unsigned)PW << 16);           // dim0[31:16]=0 | tensor_dim1[15:0]<<16
    g1[3] = (int)(tile_w << 16);                 // dim1[31:16]=0 | tile_dim0<<16
    g1[4] = (int)tile_h;                         // tile_dim1 | tile_dim2=0
    g1[5] = (int)stride;                         // tensor_dim0_stride[31:0]
    g1[6] = 0;                                   // stride hi | dim1_stride lo (2D)
    g1[7] = 0;
    const i32x4 z4 = (i32x4)0;
#if __clang_major__ >= 23
    __builtin_amdgcn_tensor_load_to_lds(g0, g1, z4, z4, (i32x8)0, 0);
#else
    __builtin_amdgcn_tensor_load_to_lds(g0, g1, z4, z4, 0);
#endif
}
#endif

// padded-space duplicate index for image index i (reflect border), or -1
__device__ __forceinline__ int mirror_pad(int i) {
    if (i >= 1 && i <= 2) return 2 - i;          // image rows 1,2 -> padded 1,0
    if (i >= 509 && i <= 510) return 1024 - i;   // image rows 509,510 -> padded 515,514
    return -1;
}

// ---------------- per-plane max reduction (order-invariant) ----------------
__global__ void plane_max_k(const float* __restrict__ src, float* __restrict__ dst, int npix) {
    __shared__ float red[256];
    const int p = blockIdx.x;
    const float* s = src + (size_t)p * npix;
    float m = -1e30f;
    for (int i = threadIdx.x; i < npix; i += 256) m = fmaxf(m, s[i]);
    red[threadIdx.x] = m;
    __syncthreads();
    for (int off = 128; off > 0; off >>= 1) {
        if ((int)threadIdx.x < off)
            red[threadIdx.x] = fmaxf(red[threadIdx.x], red[threadIdx.x + off]);
        __syncthreads();
    }
    if (threadIdx.x == 0) dst[p] = red[0];
}

// ---------------- Sobel + tangent rotation -> padded arrays ----------------
__global__ __launch_bounds__(256) void sobel_k(const float* __restrict__ img,
                                               const float* __restrict__ maxI,
                                               float* __restrict__ gmagP,
                                               float* __restrict__ xnP,
                                               float* __restrict__ ynP) {
    __shared__ float s[34 * 36];                 // 34x34 halo, pitch 36
    const int p = blockIdx.z;
    const int bx = blockIdx.x * 32, by = blockIdx.y * 32;
    const int tid = threadIdx.x;
    const float* ip = img + (size_t)p * HW;

    for (int idx = tid; idx < 34 * 34; idx += 256) {
        const int ly = idx / 34, lx = idx % 34;
        const int iy = by + ly - 1, ix = bx + lx - 1;
        float* dstp = &s[ly * 36 + lx];
        if (iy >= 0 && iy < H && ix >= 0 && ix < W) {
            async_g2l_b32(dstp, ip + iy * W + ix);   // ASYNCcnt DMA into LDS
        } else {
            *dstp = 0.0f;                            // conv zero padding
        }
    }
    wait_async();
    __syncthreads();

    const float inv = 1.0f / maxI[p];
    const float ct = -4.37113883e-08f;  // cosf(float32(pi/2))
    const float st = 1.0f;              // sinf(float32(pi/2))
    const size_t pbase = (size_t)p * PHW;
    const int tx = tid & 31, ty0 = tid >> 5;
    for (int r = 0; r < 4; ++r) {
        const int ry = ty0 + r * 8;
        const float* c = &s[(ry + 1) * 36 + (tx + 1)];
        const float a00 = c[-37], a01 = c[-36], a02 = c[-35];
        const float a10 = c[-1],  a12 = c[1];
        const float a20 = c[35],  a21 = c[36],  a22 = c[37];
        float gx = (a02 - a00) + 2.0f * (a12 - a10) + (a22 - a20);
        float gy = (a20 - a00) + 2.0f * (a21 - a01) + (a22 - a02);
        gx = fmaxf(gx * inv, 1e-12f);
        gy = fmaxf(gy * inv, 1e-12f);
        const float g = sqrtf(gx * gx + gy * gy);
        const float x0 = gx / g, y0 = gy / g;
        const float xv = x0 * ct - y0 * st;
        const float yv = y0 * ct + x0 * st;

        const int py = by + ry, px = bx + tx;
        int prs[2], pcs[2];
        int nr = 1, nc = 1;
        prs[0] = py + 2; pcs[0] = px + 2;
        int m = mirror_pad(py); if (m >= 0) prs[nr++] = m;
        m = mirror_pad(px);     if (m >= 0) pcs[nc++] = m;
        for (int rr = 0; rr < nr; ++rr)
            for (int cc = 0; cc < nc; ++cc) {
                const size_t o = pbase + (size_t)prs[rr] * PW + pcs[cc];
                gmagP[o] = g; xnP[o] = xv; ynP[o] = yv;
            }
    }
}

// ---------------- one ETF refinement iteration (padded in/out) -------------
__global__ __launch_bounds__(256) void etf_k(const float* __restrict__ xin,
                                             const float* __restrict__ yin,
                                             const float* __restrict__ gmagP,
                                             const float* __restrict__ maxG,
                                             float* __restrict__ xout,
                                             float* __restrict__ yout) {
    __shared__ float sg[TD * TD];
    __shared__ float sx[TD * TD];
    __shared__ float sy[TD * TD];
    const int p = blockIdx.z;
    const int bx = blockIdx.x * 32, by = blockIdx.y * 32;
    const int tid = threadIdx.x;
    const size_t pbase = (size_t)p * PHW;
    const size_t goff = pbase + (size_t)by * PW + bx;  // padded tile corner

#ifdef HAVE_TDM
    if (tid < 32) {                       // one wave issues the tile DMAs
        tdm_load_2d(sg, gmagP + goff, TD, TD, PW);
        tdm_load_2d(sx, xin   + goff, TD, TD, PW);
        tdm_load_2d(sy, yin   + goff, TD, TD, PW);
        wait_tensor();                    // TENSORcnt drains in issuing wave
    }
#else
    for (int idx = tid; idx < TD * TD; idx += 256) {
        const int ly = idx / TD, lx = idx % TD;
        const size_t go = pbase + (size_t)(by + ly) * PW + (bx + lx);
        async_g2l_b32(&sg[idx], gmagP + go);
        async_g2l_b32(&sx[idx], xin + go);
        async_g2l_b32(&sy[idx], yin + go);
    }
    wait_async();
#endif
    __syncthreads();

    const float invG = 1.0f / maxG[p];    // gnorm scale only matters inside tanh
    const int tx = tid & 31, ty0 = tid >> 5;
    for (int r = 0; r < 4; ++r) {
        const int ry = ty0 + r * 8;
        const int cc = (ry + 2) * TD + (tx + 2);
        const float xc = sx[cc], yc = sy[cc], gc = sg[cc];
        float xr = 0.0f, yr = 0.0f;
#pragma unroll
        for (int j = -2; j <= 2; ++j) {
#pragma unroll
            for (int i = -2; i <= 2; ++i) {
                const int nb = cc + j * TD + i;
                const float xnb = sx[nb], ynb = sy[nb], gnb = sg[nb];
                const float dot = xc * xnb + yc * ynb;                       // Wd*phi
                const float wm = 0.5f * (1.0f + tanhf((gnb - gc) * invG));   // Wm
                const float k = wm * dot;
                xr += gnb * xnb * k;      // gmag scale cancels in normalization
                yr += gnb * ynb * k;
            }
        }
        const float mag = sqrtf(xr * xr + yr * yr);
        const float xv = xr / mag, yv = yr / mag;

        const int py = by + ry, px = bx + tx;
        int prs[2], pcs[2];
        int nr = 1, nc = 1;
        prs[0] = py + 2; pcs[0] = px + 2;
        int m = mirror_pad(py); if (m >= 0) prs[nr++] = m;
        m = mirror_pad(px);     if (m >= 0) pcs[nc++] = m;
        for (int rr = 0; rr < nr; ++rr)
            for (int c2 = 0; c2 < nc; ++c2) {
                const size_t o = pbase + (size_t)prs[rr] * PW + pcs[c2];
                xout[o] = xv; yout[o] = yv;
            }
    }
}

// ---------------- per-plane angle stats over interior ----------------------
__global__ void stats_k(const float* __restrict__ xn, const float* __restrict__ yn,
                        float* __restrict__ mean, float* __restrict__ invstd) {
    __shared__ float rs[256], rq[256];
    const int p = blockIdx.x;
    const float* xs = xn + (size_t)p * PHW;
    const float* ys = yn + (size_t)p * PHW;
    float s = 0.0f, q = 0.0f;
    for (int i = threadIdx.x; i < HW; i += 256) {
        const int row = i >> 9, col = i & 511;
        const size_t o = (size_t)(row + 2) * PW + (col + 2);
        const float a = 57.29577951308232f * atanf(-ys[o] / xs[o]);
        s += a;
        q += a * a;
    }
    rs[threadIdx.x] = s;
    rq[threadIdx.x] = q;
    __syncthreads();
    for (int off = 128; off > 0; off >>= 1) {
        if ((int)threadIdx.x < off) {
            rs[threadIdx.x] += rs[threadIdx.x + off];
            rq[threadIdx.x] += rq[threadIdx.x + off];
        }
        __syncthreads();
    }
    if (threadIdx.x == 0) {
        const float mn = rs[0] / (float)HW;
        const float v = rq[0] / (float)HW - mn * mn;   // biased variance
        mean[p] = mn;
        invstd[p] = rsqrtf(v + 1e-5f);
    }
}

// ---------------- final instance-norm write --------------------------------
__global__ void final_k(const float* __restrict__ xn, const float* __restrict__ yn,
                        const float* __restrict__ mean, const float* __restrict__ invstd,
                        float* __restrict__ out, int total) {
    const int i = blockIdx.x * blockDim.x + threadIdx.x;
    if (i >= total) return;
    const int p = i / HW;
    const int rem = i - p * HW;
    const int row = rem >> 9, col = rem & 511;
    const size_t o = (size_t)p * PHW + (size_t)(row + 2) * PW + (col + 2);
    const float a = 57.29577951308232f * atanf(-yn[o] / xn[o]);
    out[i] = (a - mean[p]) * invstd[p];
}

extern "C" void kernel_launch(void* const* d_in, const int* in_sizes, int n_in,
                              void* d_out, int out_size, void* d_ws, size_t ws_size,
                              hipStream_t stream) {
    const float* img = (const float*)d_in[0];
    const int total = in_sizes[0];
    const int P = total / HW;              // B*C planes

    float* ws = (float*)d_ws;
    float* gmagP = ws;                              // padded 516x516 per plane
    float* xnA   = gmagP + (size_t)P * PHW;
    float* ynA   = xnA  + (size_t)P * PHW;
    float* xnB   = ynA  + (size_t)P * PHW;
    float* ynB   = xnB  + (size_t)P * PHW;
    float* maxI   = ynB + (size_t)P * PHW;
    float* maxG   = maxI + P;
    float* mean   = maxG + P;
    float* invstd = mean + P;
    float* out = (float*)d_out;

    dim3 tgrid(W / 32, H / 32, P);

    plane_max_k<<<P, 256, 0, stream>>>(img, maxI, HW);
    sobel_k<<<tgrid, 256, 0, stream>>>(img, maxI, gmagP, xnA, ynA);
    // duplicated reflect-border values cannot change a plane max
    plane_max_k<<<P, 256, 0, stream>>>(gmagP, maxG, PHW);

    etf_k<<<tgrid, 256, 0, stream>>>(xnA, ynA, gmagP, maxG, xnB, ynB);
    etf_k<<<tgrid, 256, 0, stream>>>(xnB, ynB, gmagP, maxG, xnA, ynA);
    etf_k<<<tgrid, 256, 0, stream>>>(xnA, ynA, gmagP, maxG, xnB, ynB);

    stats_k<<<P, 256, 0, stream>>>(xnB, ynB, mean, invstd);
    final_k<<<(total + 255) / 256, 256, 0, stream>>>(xnB, ynB, mean, invstd, out, total);
}